// graph_constructor_gat_1185410973773
// MI455X (gfx1250) — compile-verified
//
#include <hip/hip_runtime.h>

#define NNODES 8192
#define DIM 256
#define ALPHA 3.0f
#define NEG_SLOPE 0.2f

typedef __attribute__((ext_vector_type(2))) float v2f;
typedef __attribute__((ext_vector_type(8))) float v8f;

__device__ __forceinline__ float lrelu(float x) {
    return x > 0.0f ? x : NEG_SLOPE * x;
}

// ---------------------------------------------------------------------------
// Kernel 1: per-node projections  s1[i] = tanh(3*emb1[idx[i]]) . attw[:256]
//                                 s2[i] = tanh(3*emb2[idx[i]]) . attw[256:]
// One wave (32 lanes) per row; each lane covers 8 contiguous features.
// ---------------------------------------------------------------------------
__global__ void gat_node_scores(const int* __restrict__ idx,
                                const float* __restrict__ emb1,
                                const float* __restrict__ emb2,
                                const float* __restrict__ attw,
                                float* __restrict__ s1,
                                float* __restrict__ s2) {
    const int wave = threadIdx.x >> 5;
    const int lane = threadIdx.x & 31;
    const int row  = blockIdx.x * 8 + wave;
    const int src  = idx[row];

    const float* r1 = emb1 + (size_t)src * DIM + lane * 8;
    const float* r2 = emb2 + (size_t)src * DIM + lane * 8;
    const float* w1 = attw + lane * 8;
    const float* w2 = attw + DIM + lane * 8;

    float acc1 = 0.0f, acc2 = 0.0f;
#pragma unroll
    for (int k = 0; k < 8; ++k) {
        acc1 += tanhf(ALPHA * r1[k]) * w1[k];
        acc2 += tanhf(ALPHA * r2[k]) * w2[k];
    }
#pragma unroll
    for (int o = 16; o >= 1; o >>= 1) {
        acc1 += __shfl_xor(acc1, o, 32);
        acc2 += __shfl_xor(acc2, o, 32);
    }
    if (lane == 0) {
        s1[row] = acc1;
        s2[row] = acc2;
    }
}

// ---------------------------------------------------------------------------
// Kernel 2: global max of s2 (softmax row-max = lrelu(s1[i]+max(s2)+b) since
// leaky_relu is monotone). One block.
// ---------------------------------------------------------------------------
__global__ void gat_s2max(const float* __restrict__ s2,
                          float* __restrict__ s2max) {
    __shared__ float red[256];
    float m = -3.402823466e38f;
    for (int j = threadIdx.x; j < NNODES; j += 256) m = fmaxf(m, s2[j]);
    red[threadIdx.x] = m;
    __syncthreads();
    for (int s = 128; s > 0; s >>= 1) {
        if (threadIdx.x < s)
            red[threadIdx.x] = fmaxf(red[threadIdx.x], red[threadIdx.x + s]);
        __syncthreads();
    }
    if (threadIdx.x == 0) *s2max = red[0];
}

// ---------------------------------------------------------------------------
// Kernel 3: fused row softmax + column scatter.
// Block = 16 rows (i0..i0+15), 8 waves x 1024 columns each.
// Score tiles s1[i]+s2[j] are produced with V_WMMA_F32_16X16X4_F32:
//   A(16x4) = [ s1 | 1 | 0 | 0 ],  B(4x16) = [ 1^T ; s2^T ; 0 ; 0 ]
// D layout (32-bit C/D 16x16): VGPR r, lanes 0-15 -> row r, lanes 16-31 -> row r+8.
// ---------------------------------------------------------------------------
__global__ void gat_softmax_rows(const int* __restrict__ idx,
                                 const float* __restrict__ s1,
                                 const float* __restrict__ s2,
                                 const float* __restrict__ s2maxp,
                                 const float* __restrict__ attb,
                                 float* __restrict__ adj) {
    __shared__ float partial[8][16];
    __shared__ float denomLDS[16];

    const int wave    = threadIdx.x >> 5;
    const int lane    = threadIdx.x & 31;
    const int i0      = blockIdx.x * 16;
    const int colbase = wave * 1024;
    const int rowOff  = (lane >> 4) << 3;   // 0 for lanes 0-15, 8 for 16-31

    const float bias  = attb[0];
    const float s2mx  = *s2maxp;

    // per-lane row maxima (constant across the row)
    float mrow[8];
#pragma unroll
    for (int r = 0; r < 8; ++r)
        mrow[r] = lrelu(s1[i0 + rowOff + r] + s2mx + bias);

    // A matrix, 16x4 f32 layout: lanes 0-15 hold rows (K=0 in .x, K=1 in .y),
    // lanes 16-31 hold K=2/3 which we zero out.
    v2f a;
    a.x = (lane < 16) ? s1[i0 + lane] : 0.0f;
    a.y = (lane < 16) ? 1.0f : 0.0f;

    // ---- pass 1: denominators -------------------------------------------
    v8f acc = {};
    for (int t = 0; t < 64; ++t) {
        const int j0 = colbase + t * 16;
        const float s2v = s2[j0 + (lane & 15)];
        v2f b;                         // B: VGPR0 lanes 0-15 = K=0 row (ones),
        b.x = (lane < 16) ? 1.0f : s2v;//    lanes 16-31 = K=1 row (s2)
        b.y = 0.0f;                    // K=2,3 rows are zero
        v8f c = {};
        v8f d = __builtin_amdgcn_wmma_f32_16x16x4_f32(
            false, a, false, b, (short)0, c, false, false);
#pragma unroll
        for (int r = 0; r < 8; ++r) {
            float sc = lrelu(d[r] + bias);
            acc[r] += __expf(sc - mrow[r]);
        }
    }
    // reduce across the 16 lanes of each half-wave (xor masks 1,2,4,8 stay in-half)
#pragma unroll
    for (int o = 1; o < 16; o <<= 1) {
#pragma unroll
        for (int r = 0; r < 8; ++r) acc[r] += __shfl_xor(acc[r], o, 32);
    }
    if (lane == 0) {
#pragma unroll
        for (int r = 0; r < 8; ++r) partial[wave][r] = acc[r];
    }
    if (lane == 16) {
#pragma unroll
        for (int r = 0; r < 8; ++r) partial[wave][8 + r] = acc[r];
    }
    __syncthreads();
    if (threadIdx.x < 16) {
        float dsum = 0.0f;
#pragma unroll
        for (int w = 0; w < 8; ++w) dsum += partial[w][threadIdx.x];
        denomLDS[threadIdx.x] = dsum;
    }
    __syncthreads();

    float rinv[8];
#pragma unroll
    for (int r = 0; r < 8; ++r) rinv[r] = 1.0f / denomLDS[rowOff + r];

    // ---- pass 2: recompute tiles, normalize, scatter columns -------------
    for (int t = 0; t < 64; ++t) {
        const int j0 = colbase + t * 16;
        const float s2v = s2[j0 + (lane & 15)];
        const int   cj  = idx[j0 + (lane & 15)];
        v2f b;
        b.x = (lane < 16) ? 1.0f : s2v;
        b.y = 0.0f;
        v8f c = {};
        v8f d = __builtin_amdgcn_wmma_f32_16x16x4_f32(
            false, a, false, b, (short)0, c, false, false);
#pragma unroll
        for (int r = 0; r < 8; ++r) {
            float sc  = lrelu(d[r] + bias);
            float wgt = __expf(sc - mrow[r]) * rinv[r];
            adj[(size_t)(i0 + rowOff + r) * NNODES + cj] = wgt;
        }
    }
}

// ---------------------------------------------------------------------------
extern "C" void kernel_launch(void* const* d_in, const int* in_sizes, int n_in,
                              void* d_out, int out_size, void* d_ws, size_t ws_size,
                              hipStream_t stream) {
    const int*   idx  = (const int*)d_in[0];
    const float* emb1 = (const float*)d_in[1];
    const float* emb2 = (const float*)d_in[2];
    const float* attw = (const float*)d_in[3];
    const float* attb = (const float*)d_in[4];
    float* adj = (float*)d_out;

    float* s1    = (float*)d_ws;       // NNODES floats
    float* s2    = s1 + NNODES;        // NNODES floats
    float* s2max = s2 + NNODES;        // 1 float

    gat_node_scores<<<NNODES / 8, 256, 0, stream>>>(idx, emb1, emb2, attw, s1, s2);
    gat_s2max<<<1, 256, 0, stream>>>(s2, s2max);
    gat_softmax_rows<<<NNODES / 16, 256, 0, stream>>>(idx, s1, s2, s2max, attb, adj);
}